// DiffusionModel1d_28492813041733
// MI455X (gfx1250) — compile-verified
//
#include <hip/hip_runtime.h>
#include <stdint.h>

// Batched tridiagonal solve u = A(exp(alpha))^{-1} f * h2, h2 = 1/1023^2.
// Hybrid register-Thomas + PCR:
//   - one 128-thread block (4 wave32s) per batch row
//   - each thread eliminates an 8-row chunk in registers (no LDS traffic)
//   - 128-unknown reduced tridiagonal system solved with 7 PCR steps in LDS,
//     rows packed as float4 -> b128 LDS ops, conflict-free
//   - global->LDS staging via gfx1250 async loads (ASYNCcnt)
//   - LDS->global results via gfx1250 async stores, transposed through LDS so
//     every async b32 store instruction is a fully coalesced 128B line
// HBM traffic is single-touch (~128 MB => ~5.5 us @ 23.3 TB/s); register
// blocking keeps LDS traffic ~12x below pure PCR so neither LDS bandwidth nor
// VALU dominates.

#define N_UNK 1023   // unknowns per system (M - 1)
#define NPAD  1024   // padded system size
#define T_RED 128    // threads per block == reduced system size
#define MCH   8      // rows per thread (T_RED * MCH == NPAD)

__device__ __forceinline__ float fast_rcp(float x) {
    return __builtin_amdgcn_rcpf(x);   // v_rcp_f32 (TRANS), ~1 ulp
}
__device__ __forceinline__ uint32_t lds_off(const void* p) {
    // low 32 bits of a generic pointer derived from LDS == byte offset
    return (uint32_t)(uintptr_t)p;
}

__global__ __launch_bounds__(T_RED)
void DiffusionModel1d_hybrid_kernel(const float* __restrict__ alpha,
                                    const float* __restrict__ f_rhs,
                                    float* __restrict__ out)
{
    __shared__ __align__(16) float  sA[NPAD];        // alpha row staging
    __shared__ __align__(16) float  sF[NPAD];        // f_rhs staging (+pad)
    __shared__ __align__(16) float4 r4[2][T_RED];    // reduced rows (a,b,c,d)
    __shared__ __align__(16) float4 sEFG[T_RED];     // chunk-first (E,F,G,-)
    __shared__ float sx[T_RED + 1];                  // solved chunk-last x
    __shared__ __align__(16) float  sX[NPAD];        // scaled result staging

    const int t    = threadIdx.x;
    const int bat  = blockIdx.x;
    const int base = t * MCH;

    // ---- Stage alpha row (4 KB) async: 2 x b128 per thread -----------------
    // INST_OFFSET applies to both the global and the LDS address, so one
    // address pair serves both transfers.
    {
        uint64_t ga = (uint64_t)(uintptr_t)(alpha + (size_t)bat * NPAD + base);
        uint32_t la = lds_off(&sA[base]);
        asm volatile("global_load_async_to_lds_b128 %0, %1, off"
                     :: "v"(la), "v"(ga) : "memory");
        asm volatile("global_load_async_to_lds_b128 %0, %1, off offset:16"
                     :: "v"(la), "v"(ga) : "memory");
    }
    // ---- Stage f_rhs (1023 floats) async; careful tail for last thread ----
    {
        uint64_t gf = (uint64_t)(uintptr_t)(f_rhs + base);
        uint32_t lf = lds_off(&sF[base]);
        if (t < T_RED - 1) {
            asm volatile("global_load_async_to_lds_b128 %0, %1, off"
                         :: "v"(lf), "v"(gf) : "memory");
            asm volatile("global_load_async_to_lds_b128 %0, %1, off offset:16"
                         :: "v"(lf), "v"(gf) : "memory");
        } else {
            asm volatile("global_load_async_to_lds_b128 %0, %1, off"
                         :: "v"(lf), "v"(gf) : "memory");
            asm volatile("global_load_async_to_lds_b64 %0, %1, off offset:16"
                         :: "v"(lf), "v"(gf) : "memory");
            asm volatile("global_load_async_to_lds_b32 %0, %1, off offset:24"
                         :: "v"(lf), "v"(gf) : "memory");
            sF[N_UNK] = 0.0f;                 // padding row rhs
        }
    }
    asm volatile("s_wait_asynccnt 0" ::: "memory");
    __syncthreads();

    // ---- K = exp(alpha) for this chunk (vectorized LDS reads) --------------
    float Kv[MCH + 1];
    {
        float4 a0 = *(const float4*)&sA[base];
        float4 a1 = *(const float4*)&sA[base + 4];
        int i8 = base + MCH; if (i8 > NPAD - 1) i8 = NPAD - 1;
        float a8 = sA[i8];
        Kv[0] = __expf(a0.x); Kv[1] = __expf(a0.y);
        Kv[2] = __expf(a0.z); Kv[3] = __expf(a0.w);
        Kv[4] = __expf(a1.x); Kv[5] = __expf(a1.y);
        Kv[6] = __expf(a1.z); Kv[7] = __expf(a1.w);
        Kv[8] = __expf(a8);
    }
    float fv[MCH];
    {
        float4 f0 = *(const float4*)&sF[base];
        float4 f1 = *(const float4*)&sF[base + 4];
        fv[0] = f0.x; fv[1] = f0.y; fv[2] = f0.z; fv[3] = f0.w;
        fv[4] = f1.x; fv[5] = f1.y; fv[6] = f1.z; fv[7] = f1.w;
    }

    // ---- Build rows + forward register sweep -------------------------------
    // Row r: a_r x_{r-1} + b_r x_r + c_r x_{r+1} = f_r with
    //   a_r = -K_r (r>0), c_r = -K_{r+1} (r<n-1),
    //   b_0 = 2K_0+K_1, b_r = K_r+K_{r+1}, b_{n-1} = K_{n-1};
    // padding row (r==1023): identity x = 0.
    // Sweep yields: x_j = dd_j - aa_j * x_{f-1} - cc_j * x_{j+1}.
    float aa[MCH], cc[MCH], dd[MCH];
    #pragma unroll
    for (int i = 0; i < MCH; ++i) {
        const int r = base + i;
        float a, b, c;
        if (r == 0)              { a = 0.0f;   b = 2.0f * Kv[i] + Kv[i+1]; c = -Kv[i+1]; }
        else if (r < N_UNK - 1)  { a = -Kv[i]; b = Kv[i] + Kv[i+1];        c = -Kv[i+1]; }
        else if (r == N_UNK - 1) { a = -Kv[i]; b = Kv[i];                  c = 0.0f;     }
        else                     { a = 0.0f;   b = 1.0f;                   c = 0.0f;     }
        const float d = (r < N_UNK) ? fv[i] : 0.0f;
        if (i == 0) {
            float rb_ = fast_rcp(b);
            aa[0] = a * rb_; cc[0] = c * rb_; dd[0] = d * rb_;
        } else {
            float rb_ = fast_rcp(b - a * cc[i-1]);
            dd[i] = rb_ * (d - a * dd[i-1]);
            aa[i] = -(rb_ * a) * aa[i-1];
            cc[i] = rb_ * c;
        }
    }

    // ---- Backward composition: x_first = E - F*x_{f-1} - G*x_last ----------
    float E = dd[MCH-2], F = aa[MCH-2], G = cc[MCH-2];
    #pragma unroll
    for (int j = MCH - 3; j >= 0; --j) {
        E = dd[j] - cc[j] * E;
        F = aa[j] - cc[j] * F;
        G = -cc[j] * G;
    }
    sEFG[t] = make_float4(E, F, G, 0.0f);
    __syncthreads();

    // ---- Reduced tridiagonal row over chunk-last unknowns y_t = x_{l_t} ----
    // x_{l_t} = dd_L - aa_L y_{t-1} - cc_L x_{f_{t+1}},
    // x_{f_{t+1}} = E' - F' y_t - G' y_{t+1}   (primes: thread t+1)
    {
        float4 nx = (t < T_RED - 1) ? sEFG[t + 1]
                                    : make_float4(0.0f, 0.0f, 0.0f, 0.0f);
        float4 row;
        row.x = aa[MCH-1];
        row.y = 1.0f - cc[MCH-1] * nx.y;
        row.z = cc[MCH-1] * nx.z;
        row.w = dd[MCH-1] - cc[MCH-1] * nx.x;
        r4[0][t] = row;
    }
    __syncthreads();

    // ---- PCR on the 128-unknown reduced system: 7 steps, b128 LDS ops ------
    int p = 0;
    #pragma unroll
    for (int s = 1; s < T_RED; s <<= 1) {
        float4 me = r4[p][t];
        float4 lo = (t - s >= 0)    ? r4[p][t - s]
                                    : make_float4(0.0f, 1.0f, 0.0f, 0.0f);
        float4 hi = (t + s < T_RED) ? r4[p][t + s]
                                    : make_float4(0.0f, 1.0f, 0.0f, 0.0f);
        float ral = -me.x * fast_rcp(lo.y);
        float rga = -me.z * fast_rcp(hi.y);
        float4 nw;
        nw.x = ral * lo.x;
        nw.y = me.y + ral * lo.z + rga * hi.x;
        nw.z = rga * hi.z;
        nw.w = me.w + ral * lo.w + rga * hi.w;
        const int q = p ^ 1;
        r4[q][t] = nw;
        __syncthreads();
        p = q;
    }

    // ---- Solve decoupled reduced unknowns and broadcast --------------------
    {
        float4 fin = r4[p][t];
        sx[t + 1] = fin.w * fast_rcp(fin.y);
    }
    if (t == 0) sx[0] = 0.0f;                  // x_{-1} (multiplied by aa==0)
    __syncthreads();

    // ---- Back substitution within chunk (registers) ------------------------
    const float xprev = sx[t];
    float xv[MCH];
    xv[MCH-1] = sx[t + 1];
    #pragma unroll
    for (int j = MCH - 2; j >= 0; --j)
        xv[j] = dd[j] - aa[j] * xprev - cc[j] * xv[j+1];

    // ---- Stage scaled results in LDS, then async-store coalesced lines -----
    const float h2 = 1.0f / ((float)N_UNK * (float)N_UNK);
    *(float4*)&sX[base]     = make_float4(h2*xv[0], h2*xv[1], h2*xv[2], h2*xv[3]);
    *(float4*)&sX[base + 4] = make_float4(h2*xv[4], h2*xv[5], h2*xv[6], h2*xv[7]);
    __syncthreads();   // all sX writes visible before the async engine reads

    // Transposed drain: lane t streams elements t + 128k. Each async b32
    // store instruction covers one contiguous 128B span of
    // `out` (1023 is odd, so wider vectors would be misaligned). The shared
    // INST_OFFSET advances the LDS source and global dest together.
    {
        uint64_t go = (uint64_t)(uintptr_t)(out + (size_t)bat * N_UNK + t);
        uint32_t lo = lds_off(&sX[t]);
        asm volatile("global_store_async_from_lds_b32 %1, %0, off"
                     :: "v"(lo), "v"(go) : "memory");
        asm volatile("global_store_async_from_lds_b32 %1, %0, off offset:512"
                     :: "v"(lo), "v"(go) : "memory");
        asm volatile("global_store_async_from_lds_b32 %1, %0, off offset:1024"
                     :: "v"(lo), "v"(go) : "memory");
        asm volatile("global_store_async_from_lds_b32 %1, %0, off offset:1536"
                     :: "v"(lo), "v"(go) : "memory");
        asm volatile("global_store_async_from_lds_b32 %1, %0, off offset:2048"
                     :: "v"(lo), "v"(go) : "memory");
        asm volatile("global_store_async_from_lds_b32 %1, %0, off offset:2560"
                     :: "v"(lo), "v"(go) : "memory");
        asm volatile("global_store_async_from_lds_b32 %1, %0, off offset:3072"
                     :: "v"(lo), "v"(go) : "memory");
        if (t < T_RED - 1) {   // element t + 896 < 1023
            asm volatile("global_store_async_from_lds_b32 %1, %0, off offset:3584"
                         :: "v"(lo), "v"(go) : "memory");
        }
    }
    // S_ENDPGM performs an implicit S_WAIT_IDLE, draining ASYNCcnt before the
    // wave retires, so the stores are guaranteed visible at kernel completion.
}

extern "C" void kernel_launch(void* const* d_in, const int* in_sizes, int n_in,
                              void* d_out, int out_size, void* d_ws, size_t ws_size,
                              hipStream_t stream) {
    const float* alpha = (const float*)d_in[0];   // [B, M] fp32
    const float* f_rhs = (const float*)d_in[1];   // [M-1]  fp32
    float* out = (float*)d_out;                   // [B, M-1] fp32

    const int M = in_sizes[1] + 1;                // 1024 grid points
    const int B = in_sizes[0] / M;                // 16384 batches
    (void)n_in; (void)out_size; (void)d_ws; (void)ws_size;

    DiffusionModel1d_hybrid_kernel<<<B, T_RED, 0, stream>>>(alpha, f_rhs, out);
}